// ShardedCXLEmbedding_25683904430110
// MI455X (gfx1250) — compile-verified
//
#include <hip/hip_runtime.h>

// Embedding gather: out[row, :] = table[indices[row], :]
//   indices: [16384*50] int32, values in [0, 1M)
//   table:   [1M, 64] float32 (256 B per row)
//   out:     [819200, 64] float32
//
// MI455X plan: pure HBM-bandwidth problem (~353 MB moved, ~15 us floor at 23.3 TB/s).
//  - gathered row reads staged global->LDS with CDNA5 async-LDS B128 ops
//    (ASYNCcnt, double-buffered 2 KB per wave, s_wait_asynccnt ordering)
//  - output written with non-temporal B128 stores so the 210 MB output
//    stream does not evict the ~143 MB unique-row working set from the
//    192 MB L2 (repeat rows then hit L2 instead of HBM)
//  - per-lane contiguous b128 index loads (1 VMEM op / 8-row group)
//  - index stream prefetched (global_prefetch_b8)
//
// Requires nRows % 8 == 0 (true for the fixed 16384*50 shape).

typedef __attribute__((ext_vector_type(4))) float v4f;
typedef __attribute__((ext_vector_type(4))) int   v4i;

#define WAVES_PER_BLOCK 8
#define THREADS (WAVES_PER_BLOCK * 32)
#define ROWS_PER_GROUP 8                    // rows gathered per wave-iteration
#define ROW_BYTES 256                       // 64 f32
#define GROUP_BYTES (ROWS_PER_GROUP * ROW_BYTES)   // 2048 B
#define GROUP_F4 (GROUP_BYTES / 16)                // 128 v4f
#define NBUF 2

__global__ __launch_bounds__(THREADS)
void emb_gather_async(const int* __restrict__ idx,
                      const float* __restrict__ table,
                      v4f* __restrict__ out,
                      int nGroups)   // nRows / 8
{
    // [NBUF][WAVES_PER_BLOCK][GROUP_F4] : 32 KB static LDS.
    __shared__ v4f smem[NBUF * WAVES_PER_BLOCK * GROUP_F4];

    // Low 32 bits of the flat shared-aperture pointer == LDS byte address
    // (ISA 10.2, LDS aperture). Feeding this
    // ptrtoint into the asm also ESCAPES the LDS global, so the asm
    // "memory" clobbers keep the drain's ds_load_b128s alive.
    const uint32_t ldsOrigin = (uint32_t)(uintptr_t)(void*)smem;

    const int lane = (int)(threadIdx.x & 31u);
    const int wave = (int)(threadIdx.x >> 5);
    const int gwave = (int)blockIdx.x * WAVES_PER_BLOCK + wave;
    const int totalWaves = (int)gridDim.x * WAVES_PER_BLOCK;

    const int h   = lane >> 4;          // 0: rows 0..3 of group, 1: rows 4..7
    const int seg = lane & 15;          // 16B segment within a 256B row

    const uint32_t waveBase  = (uint32_t)wave * (uint32_t)GROUP_BYTES;
    const uint32_t bufStride = (uint32_t)WAVES_PER_BLOCK * (uint32_t)GROUP_BYTES;

    // Issue 4 async B128 gather loads (global -> LDS) for row-group g into buffer b.
    // Async instr j covers rows {j, 4+j}: lane -> rowLocal = h*4 + j, segment = seg.
    // Hence lane L's 4 indices are idx[rowBase + h*4 .. +3] -> one b128 load.
    auto issue = [&](int g, int b) {
        const int rowBase = g * ROWS_PER_GROUP;
        const uint32_t ldsBase = ldsOrigin + (uint32_t)b * bufStride + waveBase
                               + (uint32_t)(h * 4) * (uint32_t)ROW_BYTES
                               + (uint32_t)seg * 16u;
        const v4i e4 = *(const v4i*)(idx + rowBase + h * 4);   // 16B-aligned
#pragma unroll
        for (int j = 0; j < 4; ++j) {
            const uint32_t voff = (uint32_t)e4[j] * (uint32_t)ROW_BYTES
                                + (uint32_t)seg * 16u;
            const uint32_t lds  = ldsBase + (uint32_t)j * (uint32_t)ROW_BYTES;
            // GVS mode: mem = SADDR(64b table base) + VADDR(u32 byte offset)
            asm volatile("global_load_async_to_lds_b128 %0, %1, %2"
                         :: "v"(lds), "v"(voff), "s"(table)
                         : "memory");
        }
    };

    // Drain buffer b (LDS, contiguous row-major) to output with NT B128 stores.
    auto drain = [&](int g, int b) {
        const uint32_t fb = ((uint32_t)b * bufStride + waveBase) >> 4;  // v4f idx
        const long obase  = (long)g * GROUP_F4;                         // 128 v4f/group
#pragma unroll
        for (int k = 0; k < 4; ++k) {
            v4f v = smem[fb + (uint32_t)k * 32u + (uint32_t)lane];
            __builtin_nontemporal_store(v, &out[obase + (long)k * 32 + lane]);
        }
    };

    int g = gwave;
    int cur = 0;
    if (g < nGroups)
        issue(g, cur);

    for (; g < nGroups; g += totalWaves) {
        const int gn = g + totalWaves;
        if (gn < nGroups) {
            // prefetch indices two iterations ahead
            __builtin_prefetch(idx + (long)(gn + totalWaves) * ROWS_PER_GROUP, 0, 0);
            issue(gn, cur ^ 1);
            // async loads complete in order: <=4 outstanding means the current
            // buffer's 4 loads (older) have landed in LDS.
            asm volatile("s_wait_asynccnt 0x4" ::: "memory");
        } else {
            asm volatile("s_wait_asynccnt 0x0" ::: "memory");
        }
        drain(g, cur);
        cur ^= 1;
    }
}

extern "C" void kernel_launch(void* const* d_in, const int* in_sizes, int n_in,
                              void* d_out, int out_size, void* d_ws, size_t ws_size,
                              hipStream_t stream) {
    const int*   idx   = (const int*)d_in[0];    // [16384*50] indices
    const float* table = (const float*)d_in[1];  // [1M, 64]
    v4f*         out   = (v4f*)d_out;            // [819200, 64] f32 -> v4f*

    const int nRows   = in_sizes[0];             // 819200 (multiple of 8)
    const int nGroups = nRows / ROWS_PER_GROUP;  // 102400

    // 1280 blocks x 8 waves = 10240 waves; 102400 groups -> 10 iters/wave.
    const int blocks = 1280;
    hipLaunchKernelGGL(emb_gather_async, dim3(blocks), dim3(THREADS), 0, stream,
                       idx, table, out, nGroups);
}